// adaptive_pixel_intensity_loss_74285754352026
// MI455X (gfx1250) — compile-verified
//
#include <hip/hip_runtime.h>
#include <math.h>

// Problem constants (reference: pred/mask are [32,1,512,512] fp32, scalar out)
static constexpr int BATCH = 32;
static constexpr int HH    = 512;
static constexpr int WW    = 512;
static constexpr int HWPX  = HH * WW;
static constexpr int TW    = 128;          // columns per block (== blockDim.x)
static constexpr int TR    = 128;          // output rows per block
static constexpr int NSX   = WW / TW;      // 4 column strips
static constexpr int NSY   = HH / TR;      // 4 row segments
static constexpr int NITER = TR + 30;      // streaming iterations per block

// ---- CDNA5 async global->LDS copy (ASYNCcnt path), GV addressing mode ----
__device__ __forceinline__ void async_ld_f32(const float* gsrc, const float* ldst)
{
    unsigned           lds = (unsigned)(uintptr_t)(const void*)ldst; // low 32 bits of flat = LDS offset
    unsigned long long ga  = (unsigned long long)(uintptr_t)(const void*)gsrc;
    asm volatile("global_load_async_to_lds_b32 %0, %1, off"
                 :: "v"(lds), "v"(ga) : "memory");
}
__device__ __forceinline__ void wait_async0()
{
    asm volatile("s_wait_asynccnt 0x0" ::: "memory");
}

// Fused kernel: horizontal box sums via hierarchy H3 -> H15 -> H31 in LDS,
// vertical box sums via sliding add/sub rings in LDS, omega + all reductions
// in one pass. Mask rows are staged with double-buffered async-to-LDS loads.
__global__ __launch_bounds__(TW) void apil_main(const float* __restrict__ pred,
                                                const float* __restrict__ mask,
                                                float* __restrict__ ws)
{
    const int tid = threadIdx.x;             // 0..127, one column per thread
    const int img = blockIdx.z;
    const int c0  = blockIdx.x * TW;
    const int r0  = blockIdx.y * TR;
    const int col = c0 + tid;

    // Double-buffered row staging + extended horizontal sums
    __shared__ float m_rows[2][160];         // column j = c0-16+l
    __shared__ float h3e [160];
    __shared__ float h15e[160];
    // Vertical sliding-window rings (per column)
    __shared__ float ring3 [3  * TW];
    __shared__ float ring15[15 * TW];
    __shared__ float ring31[31 * TW];
    // Delay rings so avg3/avg15/avg31 align at the same output row
    __shared__ float dly3 [15 * TW];
    __shared__ float dly15[9  * TW];
    __shared__ float red[TW];

    for (int i = tid; i < 3  * TW; i += TW) ring3 [i] = 0.0f;
    for (int i = tid; i < 15 * TW; i += TW) ring15[i] = 0.0f;
    for (int i = tid; i < 31 * TW; i += TW) ring31[i] = 0.0f;

    const float* mimg = mask + (size_t)img * HWPX;
    const float* pimg = pred + (size_t)img * HWPX;

    // Issue async staging of row (r0-15+uu) into m_rows[uu&1].
    // Every lane issues exactly 2 loads -> uniform ASYNCcnt per wave.
    // Slot 128+(tid&31) is written redundantly by all 4 waves (same data).
    auto issue_row = [&](int uu) {
        const int iru = r0 - 15 + uu;
        if (iru < 0 || iru >= HH) return;                 // uniform
        const float* mrow = mimg + (size_t)iru * WW;
        float* dst = m_rows[uu & 1];
        int jg = c0 - 16 + tid;
        int jc = min(max(jg, 0), WW - 1);                 // clamp, fix up later
        async_ld_f32(mrow + jc, dst + tid);
        int i1  = 128 + (tid & 31);
        int j1  = c0 - 16 + i1;                           // >= 112, can only overflow
        int j1c = min(j1, WW - 1);
        async_ld_f32(mrow + j1c, dst + i1);
    };

    issue_row(0);                                         // prime the pipeline

    float A3 = 0.0f, A15 = 0.0f, A31 = 0.0f;
    float s_om = 0.0f, s_in = 0.0f, s_un = 0.0f, s_bce = 0.0f, s_mae = 0.0f;

    // ir = r0-15 .. r0+TR+14 ; finalize row r = ir-15 for r in [r0, r0+TR)
    for (int u = 0; u < NITER; ++u) {
        const int  ir = r0 - 15 + u;
        const bool rv = (ir >= 0) && (ir < HH);
        float* mb = m_rows[u & 1];

        wait_async0();                      // row u has landed in LDS
        // zero-padding fixups (issuing wave's own slots, after its wait)
        if (rv) {
            int jg = c0 - 16 + tid;
            if (jg < 0 || jg >= WW) mb[tid] = 0.0f;
            int i1 = 128 + (tid & 31);
            if (c0 - 16 + i1 >= WW) mb[i1] = 0.0f;
        } else {
            mb[tid] = 0.0f;
            mb[128 + (tid & 31)] = 0.0f;
        }
        __syncthreads();                    // staging of row u visible to all
        if (u + 1 < NITER) issue_row(u + 1);   // overlap next row with compute
        if (rv)                              // pred row used at finalize (u+15)
            __builtin_prefetch(pimg + (size_t)ir * WW + col, 0, 0);

        // ---- H3 over l in [1,158] : window of 3 ----
        {
            int l = 1 + tid;
            h3e[l] = mb[l - 1] + mb[l] + mb[l + 1];
            if (tid < 30) {
                int l2 = 129 + tid;
                h3e[l2] = mb[l2 - 1] + mb[l2] + mb[l2 + 1];
            }
        }
        __syncthreads();
        // ---- H15 over l in [8,151] : five disjoint H3 windows ----
        {
            int l = 8 + tid;
            h15e[l] = h3e[l - 6] + h3e[l - 3] + h3e[l] + h3e[l + 3] + h3e[l + 6];
            if (tid < 16) {
                int l2 = 136 + tid;
                h15e[l2] = h3e[l2 - 6] + h3e[l2 - 3] + h3e[l2] + h3e[l2 + 3] + h3e[l2 + 6];
            }
        }
        __syncthreads();

        const int   lc   = tid + 16;
        const float h3c  = h3e[lc];
        const float h15c = h15e[lc];
        const float h31c = h15e[lc - 8] + h15e[lc + 8] + mb[lc];

        // ---- vertical sliding windows (thread-private column -> no sync) ----
        const int s3 = u % 3, s15 = u % 15, s31 = u % 31, s9 = u % 9;
        float o;
        o = ring3 [s3  * TW + tid]; A3  += h3c  - o; ring3 [s3  * TW + tid] = h3c;
        o = ring15[s15 * TW + tid]; A15 += h15c - o; ring15[s15 * TW + tid] = h15c;
        o = ring31[s31 * TW + tid]; A31 += h31c - o; ring31[s31 * TW + tid] = h31c;

        // A3 now = window for row ir-1, A15 for row ir-7, A31 for row ir-15
        dly3 [s15 * TW + tid] = A3  * (1.0f / 9.0f);
        dly15[s9  * TW + tid] = A15 * (1.0f / 225.0f);

        const int r = ir - 15;
        if (r >= r0 && r < r0 + TR) {           // uniform across the block
            const float a31 = A31 * (1.0f / 961.0f);
            const float a3  = dly3 [((u + 1) % 15) * TW + tid]; // written at u-14
            const float a15 = dly15[((u + 1) % 9)  * TW + tid]; // written at u-8
            const size_t off = (size_t)r * WW + col;
            const float m = mimg[off];
            const float p = pimg[off];
            const float w   = fabsf(a3 - m) + fabsf(a15 - m) + fabsf(a31 - m);
            const float omg = 1.0f + 0.5f * w * m;
            s_om += omg;
            s_in += p * m * omg;
            s_un += (p + m) * omg;
            const float lp  = fmaxf(logf(p),    -100.0f);   // torch log clamp
            const float l1p = fmaxf(log1pf(-p), -100.0f);
            s_bce += -(m * lp + (1.0f - m) * l1p);
            s_mae += fabsf(p - m);
        }
    }

    // ---- deterministic block reduction: 5 partials -> ws[flat*5 + q] ----
    const int flat = (blockIdx.z * NSY + blockIdx.y) * NSX + blockIdx.x;
    float vals[5] = { s_om, s_in, s_un, s_bce, s_mae };
    for (int q = 0; q < 5; ++q) {
        __syncthreads();
        red[tid] = vals[q];
        __syncthreads();
        for (int offr = TW / 2; offr > 0; offr >>= 1) {
            if (tid < offr) red[tid] += red[tid + offr];
            __syncthreads();
        }
        if (tid == 0) ws[flat * 5 + q] = red[0];
    }
}

// Epilogue: fold 512 block-partials into per-image sums, global bce/mae,
// then the scalar loss. One wave32, fixed order -> deterministic.
__global__ __launch_bounds__(32) void apil_final(const float* __restrict__ ws,
                                                 float* __restrict__ out)
{
    __shared__ float sb[BATCH], sm[BATCH], st[BATCH];
    const int b = threadIdx.x;          // 0..31 == image index
    float om = 0.0f, in_ = 0.0f, un = 0.0f, bce = 0.0f, mae = 0.0f;
    for (int k = 0; k < NSX * NSY; ++k) {
        const float* e = ws + ((size_t)b * (NSX * NSY) + k) * 5;
        om  += e[0]; in_ += e[1]; un  += e[2]; bce += e[3]; mae += e[4];
    }
    sb[b] = bce; sm[b] = mae;
    __syncthreads();
    float tb = 0.0f, tm = 0.0f;
    for (int i = 0; i < BATCH; ++i) { tb += sb[i]; tm += sm[i]; }
    const float nall  = (float)BATCH * (float)HWPX;     // 8388608, exact
    const float bceM  = tb / nall;
    const float maeM  = tm / nall;
    const float abce  = bceM * om / (om + 0.5f * (float)HWPX);
    const float amae  = maeM * om / (om - (float)HWPX);
    const float aiou  = 1.0f - (in_ + 1.0f) / (un - in_ + 1.0f);
    st[b] = 0.7f * (abce + aiou + amae);
    __syncthreads();
    if (b == 0) {
        float t = 0.0f;
        for (int i = 0; i < BATCH; ++i) t += st[i];
        out[0] = t / (float)BATCH;
    }
}

extern "C" void kernel_launch(void* const* d_in, const int* in_sizes, int n_in,
                              void* d_out, int out_size, void* d_ws, size_t ws_size,
                              hipStream_t stream)
{
    const float* pred = (const float*)d_in[0];
    const float* mask = (const float*)d_in[1];
    float* out = (float*)d_out;
    float* ws  = (float*)d_ws;   // needs BATCH*NSX*NSY*5 floats = 10 KB

    dim3 grid(NSX, NSY, BATCH);
    apil_main <<<grid, dim3(TW), 0, stream>>>(pred, mask, ws);
    apil_final<<<1,    dim3(32), 0, stream>>>(ws, out);
}